// Correlation_15719580303978
// MI455X (gfx1250) — compile-verified
//
#include <hip/hip_runtime.h>

// FlowNet correlation cost volume on MI455X (gfx1250), fp32 WMMA, register-resident fragments.
//
// out[b, dy*21+dx, y, x] = (s1*s2/(96*os)) * sum_c d1[b,c,y,x]*d2[b,c,y+2dy-20,x+2dx-20]
//
// Per workgroup: one (b, y, 16-wide x tile). For each of 21 dy values, a banded
// GEMM P = A(16x96) x B(96x64) via V_WMMA_F32_16X16X4_F32, band-extract P[m, m+2j].
// A fragments (dy-invariant) live in 48 VGPRs for the whole block; B fragments are
// loaded straight from L2-resident d2 into 48 VGPRs per dy (no LDS staging: each B
// element feeds exactly one WMMA k-step of one wave -- the 16-row reuse happens
// inside the matrix unit). LDS is used only to transpose the band for coalesced stores.

typedef __attribute__((ext_vector_type(2))) float v2f;
typedef __attribute__((ext_vector_type(8))) float v8f;

#define B_DIM   8
#define C_DIM   96
#define H_DIM   64
#define W_DIM   96
#define MD      20
#define DSP     21                  // displacements per axis
#define NDISP   (DSP * DSP)         // 441
#define TILE_M  16
#define KSTEPS  (C_DIM / 4)         // 24 WMMA K-steps
#define CH_STR  (H_DIM * W_DIM)     // channel stride in elements

__global__ __launch_bounds__(256)
void corr_wmma_f32_kernel(const float* __restrict__ d1,
                          const float* __restrict__ d2,
                          const float* __restrict__ s1,
                          const float* __restrict__ s2,
                          const float* __restrict__ os,
                          float* __restrict__ out) {
    __shared__ float ldsOut[2][DSP * TILE_M];   // 2.7 KB band staging [group][j][m]

    const int x0   = blockIdx.x * TILE_M;
    const int y    = blockIdx.y;
    const int b    = blockIdx.z;
    const int tid  = threadIdx.x;
    const int g    = tid >> 7;          // dy-group (0/1): waves 0-3 vs 4-7
    const int tg   = tid & 127;         // thread id within group
    const int wl   = tg >> 5;           // wave within group -> N-tile index
    const int lane = tid & 31;
    const int lm   = lane & 15;
    const int hi   = lane >> 4;         // lane half: K offset +2 (A/B), M offset +8 (C/D)
    const int nbase = wl * 16;

    const float comb = s1[0] * s2[0] / (96.0f * os[0]);

    const float* d1b = d1 + (size_t)b * C_DIM * CH_STR;
    const float* d2b = d2 + (size_t)b * C_DIM * CH_STR;

    // ---- Preload A fragments for the whole block (dy-invariant): 48 VGPRs.
    // ISA 16x4 f32 A layout: lane holds row m=lm, K pair {kk*4+2*hi, +1}.
    v2f af[KSTEPS];
    {
        const float* ap = d1b + (size_t)(hi * 2) * CH_STR + y * W_DIM + x0 + lm;
#pragma unroll
        for (int kk = 0; kk < KSTEPS; ++kk) {
            af[kk].x = ap[(size_t)(kk * 4    ) * CH_STR];
            af[kk].y = ap[(size_t)(kk * 4 + 1) * CH_STR];
        }
    }

    // This wave's absolute B column for the banded GEMM (per-lane, may be OOB -> 0).
    const int  xx    = x0 - MD + nbase + lm;
    const bool colOK = ((unsigned)xx < (unsigned)W_DIM);

    for (int itp = 0; itp < (DSP + 1) / 2; ++itp) {   // 11 iterations, 2 dy per iter
        const int  dyIdx  = itp * 2 + g;              // uniform per wave
        const bool active = (dyIdx < DSP);

        if (active) {
            const int  yy    = y + dyIdx * 2 - MD;
            const bool rowOK = ((unsigned)yy < (unsigned)H_DIM);

            // ---- B fragments straight from global (L2-resident), zero-padded.
            v2f bf[KSTEPS];
#pragma unroll
            for (int kk = 0; kk < KSTEPS; ++kk) { bf[kk].x = 0.0f; bf[kk].y = 0.0f; }
            if (rowOK && colOK) {   // per-lane predication: loads only, WMMA is outside
                const float* bp = d2b + (size_t)(hi * 2) * CH_STR + yy * W_DIM + xx;
#pragma unroll
                for (int kk = 0; kk < KSTEPS; ++kk) {
                    bf[kk].x = bp[(size_t)(kk * 4    ) * CH_STR];
                    bf[kk].y = bp[(size_t)(kk * 4 + 1) * CH_STR];
                }
            }

            // ---- 24 WMMAs, two independent accumulator chains for XDL ILP.
            v8f acc0 = {}, acc1 = {};
#pragma unroll
            for (int kk = 0; kk < KSTEPS; kk += 2) {
                acc0 = __builtin_amdgcn_wmma_f32_16x16x4_f32(
                        false, af[kk],     false, bf[kk],     (short)0, acc0, false, false);
                acc1 = __builtin_amdgcn_wmma_f32_16x16x4_f32(
                        false, af[kk + 1], false, bf[kk + 1], (short)0, acc1, false, false);
            }
            const v8f acc = acc0 + acc1;

            // ---- Band extraction: P[m, col] needed iff col-m even, (col-m)/2 in [0,21).
            // C/D layout (ISA 7.12.2): VGPR r -> M = r + 8*hi, N = lane&15.
#pragma unroll
            for (int r = 0; r < 8; ++r) {
                const int m   = r + 8 * hi;
                const int col = nbase + lm;
                const int rel = col - m;
                if (rel >= 0 && !(rel & 1)) {
                    const int j = rel >> 1;
                    if (j < DSP) ldsOut[g][j * TILE_M + m] = acc[r];
                }
            }
        }
        __syncthreads();

        // ---- Coalesced store: 21 displacement channels x 16 contiguous floats.
        if (active) {
            for (int idx = tg; idx < DSP * TILE_M; idx += 128) {
                const int j = idx >> 4, m = idx & 15;
                const int d = dyIdx * DSP + j;
                out[(((size_t)b * NDISP + d) * H_DIM + y) * W_DIM + x0 + m] =
                    ldsOut[g][idx] * comb;
            }
        }
        __syncthreads();
    }
}

extern "C" void kernel_launch(void* const* d_in, const int* in_sizes, int n_in,
                              void* d_out, int out_size, void* d_ws, size_t ws_size,
                              hipStream_t stream) {
    const float* d1 = (const float*)d_in[0];
    const float* d2 = (const float*)d_in[1];
    const float* s1 = (const float*)d_in[2];
    const float* s2 = (const float*)d_in[3];
    // d_in[4] = inter_scale (unused by reference math)
    const float* os = (const float*)d_in[5];

    dim3 grid(W_DIM / TILE_M, H_DIM, B_DIM);   // 6 x 64 x 8 = 3072 workgroups
    corr_wmma_f32_kernel<<<grid, 256, 0, stream>>>(d1, d2, s1, s2, os, (float*)d_out);
}